// ProjectLayer_11673721111205
// MI455X (gfx1250) — compile-verified
//
#include <hip/hip_runtime.h>
#include <hip/hip_bf16.h>

// ---------------- problem constants (fixed by the reference) ----------------
constexpr int   Bb  = 4;      // batches
constexpr int   Nv  = 5;      // views
constexpr int   Jj  = 15;     // joints
constexpr int   Hh  = 128;    // heatmap H
constexpr int   Wh  = 240;    // heatmap W
constexpr int   CBX = 80, CBY = 80, CBZ = 20;
constexpr int   P   = CBX * CBY * CBZ;          // 128000 voxels
constexpr float IMG_W = 960.0f, IMG_H = 512.0f;
constexpr float GSX = 8000.0f, GSY = 8000.0f, GSZ = 2000.0f;

// sg-clip(+/-1.1) round-trip collapses to a direct clamp in heatmap coords:
// gx = (clip(ax/(W-1)*2-1, -1.1, 1.1)+1)*0.5*(W-1) == clip(ax, -.05(W-1), 1.05(W-1))
constexpr float GXLO = -0.05f * (float)(Wh - 1);   // -11.95
constexpr float GXHI =  1.05f * (float)(Wh - 1);   // 250.95
constexpr float GYLO = -0.05f * (float)(Hh - 1);   //  -6.35
constexpr float GYHI =  1.05f * (float)(Hh - 1);   // 133.35

typedef __attribute__((ext_vector_type(2))) float v2f;
typedef __attribute__((ext_vector_type(8))) float v8f;

__device__ __forceinline__ int imin(int a, int b) { return a < b ? a : b; }
__device__ __forceinline__ int imax(int a, int b) { return a > b ? a : b; }

// One wave handles a tile of 16 voxels.
//  - A-matrix (batch-invariant): row m=3n+i -> [Rn_i0,Rn_i1,Rn_i2, -Rn_i.Tn]
//  - B-matrix: column c -> [x,y,z,1] of voxel c
//  - D[m,c] = camera coordinate i of view n for voxel c (15 useful rows).
// Lane c   (0..15) receives D[0..7 ,c] -> views 0,1 and view2 x,y
// Lane c+16       receives D[8..15,c] -> view2 z, views 3,4
__global__ __launch_bounds__(256) void project_fuse_kernel(
    const float* __restrict__ hm,   // [N,B,J,Hh,Wh]
    const float* __restrict__ R,    // [B,N,3,3]
    const float* __restrict__ T,    // [B,N,3]
    const float* __restrict__ F,    // [B,N,2]
    const float* __restrict__ C,    // [B,N,2]
    const float* __restrict__ TR,   // [B,N,2,3]
    const float* __restrict__ WHp,  // [B,N,2]
    const float* __restrict__ GC,   // [B,3]
    float* __restrict__ out_fused,  // [B,J,P]
    float* __restrict__ out_grids)  // [B,P,3]
{
    const int  lane   = threadIdx.x & 31;
    const int  wave   = threadIdx.x >> 5;
    const bool hiHalf = lane >= 16;
    const int  colPt  = lane & 15;

    constexpr int blocksPerBatch = P / 128;           // 1000
    const int b    = blockIdx.x / blocksPerBatch;
    const int tile = (blockIdx.x % blocksPerBatch) * 8 + wave;
    const int pt   = tile * 16 + colPt;               // this lane's voxel

    // ---------------- A operand: projection rows (uniform flow) ----------------
    const int   m      = colPt;                 // channel index 0..15
    const float validA = (m < 15) ? 1.0f : 0.0f;
    int n_a = m / 3;  n_a = imin(n_a, Nv - 1);
    const int i_a = m - 3 * (m / 3);
    const float* Rrow = R + (((size_t)b * Nv + n_a) * 3 + i_a) * 3;
    const float* Tn   = T + ((size_t)b * Nv + n_a) * 3;
    const float r0 = Rrow[0] * validA;
    const float r1 = Rrow[1] * validA;
    const float r2 = Rrow[2] * validA;
    const float rt = -(Rrow[0] * Tn[0] + Rrow[1] * Tn[1] + Rrow[2] * Tn[2]) * validA;
    v2f Aop;                       // VGPR0: K0 / K2 ; VGPR1: K1 / K3
    Aop.x = hiHalf ? r2 : r0;
    Aop.y = hiHalf ? rt : r1;

    // ---------------- B operand: voxel world positions ----------------
    const int ix  = pt / (CBY * CBZ);
    const int rem = pt - ix * (CBY * CBZ);
    const int iy  = rem / CBZ;
    const int iz  = rem - iy * CBZ;
    const float gcx = GC[b * 3 + 0], gcy = GC[b * 3 + 1], gcz = GC[b * 3 + 2];
    const float wxp = -GSX * 0.5f + (float)ix * (GSX / (float)(CBX - 1)) + gcx;
    const float wyp = -GSY * 0.5f + (float)iy * (GSY / (float)(CBY - 1)) + gcy;
    const float wzp = -GSZ * 0.5f + (float)iz * (GSZ / (float)(CBZ - 1)) + gcz;
    v2f Bop;
    Bop.x = hiHalf ? wzp : wxp;    // K0 / K2
    Bop.y = hiHalf ? 1.0f : wyp;   // K1 / K3 (homogeneous 1)

    v8f Cz = {};
    // All 5 view projections for 16 voxels in one matrix op.
    v8f Dm = __builtin_amdgcn_wmma_f32_16x16x4_f32(
        false, Aop, false, Bop, (short)0, Cz, false, false);

    // view2's cam-z lives in row 8 (lane c+16, VGPR0) -> fetch across the half.
    const float v2z = __shfl_xor(Dm[0], 16, 32);

    float camv[3][3];
    camv[0][0] = hiHalf ? Dm[1] : Dm[0];
    camv[0][1] = hiHalf ? Dm[2] : Dm[1];
    camv[0][2] = hiHalf ? Dm[3] : Dm[2];
    camv[1][0] = hiHalf ? Dm[4] : Dm[3];
    camv[1][1] = hiHalf ? Dm[5] : Dm[4];
    camv[1][2] = hiHalf ? Dm[6] : Dm[5];
    camv[2][0] = Dm[6];                 // only meaningful for low half
    camv[2][1] = Dm[7];
    camv[2][2] = v2z;

    const int nvv   = hiHalf ? 2 : 3;
    const int vbase = hiHalf ? 3 : 0;

    float acc[Jj];
#pragma unroll
    for (int j = 0; j < Jj; ++j) acc[j] = 0.0f;
    float cnt = 0.0f;

    constexpr int HW = Hh * Wh;
#pragma unroll
    for (int v = 0; v < 3; ++v) {
        if (v >= nvv) break;
        const int n = vbase + v;
        const size_t bn = (size_t)b * Nv + n;
        const float fx  = F[bn * 2 + 0],  fy  = F[bn * 2 + 1];
        const float cx0 = C[bn * 2 + 0],  cy0 = C[bn * 2 + 1];
        const float Wd  = WHp[bn * 2 + 0], Hd = WHp[bn * 2 + 1];
        const float t00 = TR[bn * 6 + 0], t01 = TR[bn * 6 + 1], t02 = TR[bn * 6 + 2];
        const float t10 = TR[bn * 6 + 3], t11 = TR[bn * 6 + 4], t12 = TR[bn * 6 + 5];

        // single v_rcp_f32 instead of the IEEE div macro sequence
        const float inv = __builtin_amdgcn_rcpf(camv[v][2]);
        float px = camv[v][0] * inv * fx + cx0;
        float py = camv[v][1] * inv * fy + cy0;

        // bound computed BEFORE clipping (matches reference)
        const bool inb = (px >= 0.0f) & (py >= 0.0f) & (px < Wd) & (py < Hd);
        if (!inb) continue;          // sample would be masked to zero anyway
        cnt += 1.0f;

        const float mx = fmaxf(Wd, Hd);
        px = fminf(fmaxf(px, -1.0f), mx);
        py = fminf(fmaxf(py, -1.0f), mx);

        float ax = t00 * px + t01 * py + t02;
        float ay = t10 * px + t11 * py + t12;
        ax *= (float)Wh / IMG_W;     // 0.25
        ay *= (float)Hh / IMG_H;     // 0.25

        // collapsed normalized-coord round trip: direct clamp in heatmap coords
        const float gx = fminf(fmaxf(ax, GXLO), GXHI);
        const float gy = fminf(fmaxf(ay, GYLO), GYHI);

        const float x0f = floorf(gx), y0f = floorf(gy);
        const float fxw = gx - x0f,  fyw = gy - y0f;
        const int x0 = (int)x0f, y0 = (int)y0f;
        const int x1 = x0 + 1,   y1 = y0 + 1;
        const float vx0 = (x0 >= 0 && x0 < Wh) ? 1.0f : 0.0f;
        const float vx1 = (x1 >= 0 && x1 < Wh) ? 1.0f : 0.0f;
        const float vy0 = (y0 >= 0 && y0 < Hh) ? 1.0f : 0.0f;
        const float vy1 = (y1 >= 0 && y1 < Hh) ? 1.0f : 0.0f;
        const int x0c = imin(imax(x0, 0), Wh - 1), x1c = imin(imax(x1, 0), Wh - 1);
        const int y0c = imin(imax(y0, 0), Hh - 1), y1c = imin(imax(y1, 0), Hh - 1);

        const float w00 = (1.0f - fxw) * (1.0f - fyw) * vx0 * vy0;
        const float w10 = fxw * (1.0f - fyw) * vx1 * vy0;
        const float w01 = (1.0f - fxw) * fyw * vx0 * vy1;
        const float w11 = fxw * fyw * vx1 * vy1;

        const int o00 = y0c * Wh + x0c, o10 = y0c * Wh + x1c;
        const int o01 = y1c * Wh + x0c, o11 = y1c * Wh + x1c;
        const float* base = hm + ((size_t)n * Bb + b) * Jj * HW;   // heatmaps[n,b,·]
#pragma unroll
        for (int j = 0; j < Jj; ++j) {
            const float* img = base + (size_t)j * HW;
            acc[j] += w00 * img[o00] + w10 * img[o10]
                    + w01 * img[o01] + w11 * img[o11];
        }
    }

    // merge the two half-wave partials (views 0-2 with views 3-4);
    // afterwards BOTH halves hold the full per-voxel sums.
    cnt += __shfl_xor(cnt, 16, 32);
#pragma unroll
    for (int j = 0; j < Jj; ++j) acc[j] += __shfl_xor(acc[j], 16, 32);

    if (pt < P) {
        const float invc = 1.0f / (cnt + 1e-6f);
        // split the 15 joint stores across the two half-waves
        const int jLo = hiHalf ? 8 : 0;
        const int jHi = hiHalf ? Jj : 8;
        for (int j = jLo; j < jHi; ++j) {
            float fz = acc[j] * invc;
            fz = fminf(fmaxf(fz, 0.0f), 1.0f);
            out_fused[((size_t)b * Jj + j) * P + pt] = fz;
        }
        if (!hiHalf) {
            float* g = out_grids + ((size_t)b * P + pt) * 3;
            g[0] = wxp; g[1] = wyp; g[2] = wzp;
        }
    }
}

extern "C" void kernel_launch(void* const* d_in, const int* in_sizes, int n_in,
                              void* d_out, int out_size, void* d_ws, size_t ws_size,
                              hipStream_t stream) {
    (void)in_sizes; (void)n_in; (void)out_size; (void)d_ws; (void)ws_size;
    const float* hm  = (const float*)d_in[0];
    const float* R   = (const float*)d_in[1];
    const float* T   = (const float*)d_in[2];
    const float* F   = (const float*)d_in[3];
    const float* C   = (const float*)d_in[4];
    const float* TR  = (const float*)d_in[5];
    const float* WHp = (const float*)d_in[6];
    const float* GC  = (const float*)d_in[7];
    float* out_fused = (float*)d_out;
    float* out_grids = out_fused + (size_t)Bb * Jj * P;

    dim3 grid(Bb * (P / 128));   // 8 waves/block, 16 voxels/wave
    dim3 block(256);
    project_fuse_kernel<<<grid, block, 0, stream>>>(hm, R, T, F, C, TR, WHp, GC,
                                                    out_fused, out_grids);
}